// BiGNN_73400991088662
// MI455X (gfx1250) — compile-verified
//
#include <hip/hip_runtime.h>

// GraphSAGE 2-layer forward for MI455X (gfx1250, wave32).
//   layer: out = (mean_{j->i} x_j) @ W_l + b_l + x_i @ W_r   [+ relu for layer 1]
// Phase A: edge scatter-add with global_atomic_add_f32 (memory-bound, ~2 GB total).
// Phase B: WMMA f32 16x16x4 GEMM, W staged in LDS, mean-normalization + bias +
//          relu fused into the GEMM (one wave = one 16x128 output tile).

typedef float v2f __attribute__((ext_vector_type(2)));
typedef float v8f __attribute__((ext_vector_type(8)));

#define FEAT 128

// ---------------------------------------------------------------- zero fill
__global__ void zero_kernel(float* __restrict__ p, long long n) {
    long long i = (long long)blockIdx.x * blockDim.x + threadIdx.x;
    long long stride = (long long)gridDim.x * blockDim.x;
    for (; i < n; i += stride) p[i] = 0.0f;
}

// ------------------------------------------------------- edge scatter (mean)
// One wave32 per edge. Each lane owns 4 consecutive floats (16B) of the
// 128-float feature row: the wave reads one contiguous 512B row of feat[src]
// and issues 4 global_atomic_add_f32 per lane into agg[dst].
__global__ void __launch_bounds__(256)
scatter_kernel(const float* __restrict__ feat,
               const int* __restrict__ src,
               const int* __restrict__ dst,
               float* __restrict__ agg,
               float* __restrict__ cnt,
               int nedges, int add_cnt) {
    int wave = (blockIdx.x * blockDim.x + threadIdx.x) >> 5;
    int lane = threadIdx.x & 31;
    if (wave >= nedges) return;

    int s = src[wave];
    int d = dst[wave];

    const float4 v = *(const float4*)(feat + (size_t)s * FEAT + lane * 4);
    float* a = agg + (size_t)d * FEAT + lane * 4;
    unsafeAtomicAdd(a + 0, v.x);
    unsafeAtomicAdd(a + 1, v.y);
    unsafeAtomicAdd(a + 2, v.z);
    unsafeAtomicAdd(a + 3, v.w);

    if (add_cnt && lane == 0) unsafeAtomicAdd(cnt + d, 1.0f);
}

// ----------------------------------------------------------------- WMMA GEMM
// Accumulate C(16x128) += rs * Arow_tile(16x128-slice) @ sW(128x128), K in
// steps of 4 using V_WMMA_F32_16X16X4_F32.
// f32 A-fragment layout: lanes 0-15 rows M=0..15; VGPR0={K+0 | K+2(hi half)},
// VGPR1={K+1 | K+3}. B-fragment mirrors it (row of B striped across lanes).
__device__ __forceinline__ void gemm_accum(const float* __restrict__ arow,
                                           float rs, bool do_scale,
                                           const float* __restrict__ sW,
                                           int lane, v8f c[8]) {
    const int kOff = (lane >> 4) << 1;   // 0 or 2
    const int colb = lane & 15;
#pragma unroll 8
    for (int kt = 0; kt < 32; ++kt) {
        const int k = kt * 4 + kOff;                 // even -> 8B aligned
        v2f a = *(const v2f*)(arow + k);
        if (do_scale) { a.x *= rs; a.y *= rs; }
#pragma unroll
        for (int j = 0; j < 8; ++j) {
            v2f b;
            b.x = sW[k * FEAT + j * 16 + colb];
            b.y = sW[(k + 1) * FEAT + j * 16 + colb];
            c[j] = __builtin_amdgcn_wmma_f32_16x16x4_f32(
                false, a, false, b, (short)0, c[j], false, false);
        }
    }
}

// out[tile rows] = (agg/max(cnt,1)) @ Wl + bias + xin @ Wr   (+ optional relu)
// 8 waves/block, one 16-row tile per wave. Wl then Wr staged through 64KB LDS.
__global__ void __launch_bounds__(256)
sage_gemm_kernel(const float* __restrict__ agg,
                 const float* __restrict__ cnt,
                 const float* __restrict__ xin,
                 const float* __restrict__ Wl,
                 const float* __restrict__ bias,
                 const float* __restrict__ Wr,
                 float* __restrict__ out,
                 int ntiles, int do_relu) {
    __shared__ float sW[FEAT * FEAT];          // 64 KB

    const int tid    = threadIdx.x;
    const int lane   = tid & 31;
    const int waveId = tid >> 5;
    const int tile   = blockIdx.x * 8 + waveId;
    const bool active = tile < ntiles;         // wave-uniform (EXEC all-1s)

    const int r0 = tile * 16;
    const int ra = r0 + (lane & 15);           // A-fragment row for this lane

    v8f c[8] = {};                             // 16x128 accumulator tile

    // ---- stage W_l, run mean @ W_l -------------------------------------
    {
        float4* d4 = (float4*)sW;
        const float4* s4 = (const float4*)Wl;
        for (int i = tid; i < FEAT * FEAT / 4; i += 256) d4[i] = s4[i];
    }
    __syncthreads();

    if (active) {
        const float rs = 1.0f / fmaxf(cnt[ra], 1.0f);
        gemm_accum(agg + (size_t)ra * FEAT, rs, true, sW, lane, c);
    }
    __syncthreads();

    // ---- stage W_r, run x @ W_r ----------------------------------------
    {
        float4* d4 = (float4*)sW;
        const float4* s4 = (const float4*)Wr;
        for (int i = tid; i < FEAT * FEAT / 4; i += 256) d4[i] = s4[i];
    }
    __syncthreads();

    if (active) {
        gemm_accum(xin + (size_t)ra * FEAT, 1.0f, false, sW, lane, c);

        // epilogue: + bias, relu, store. C layout: VGPR v, lanes 0-15 ->
        // row r0+v col=lane; lanes 16-31 -> row r0+8+v col=lane-16.
        const int colb  = lane & 15;
        const int rbase = r0 + ((lane >> 4) << 3);
#pragma unroll
        for (int j = 0; j < 8; ++j) {
            const int col = j * 16 + colb;
            const float bv = bias[col];
#pragma unroll
            for (int v = 0; v < 8; ++v) {
                float val = c[j][v] + bv;
                if (do_relu) val = fmaxf(val, 0.0f);
                out[(size_t)(rbase + v) * FEAT + col] = val;
            }
        }
    }
}

// ------------------------------------------------------------------ launch
extern "C" void kernel_launch(void* const* d_in, const int* in_sizes, int n_in,
                              void* d_out, int out_size, void* d_ws, size_t ws_size,
                              hipStream_t stream) {
    const float* x   = (const float*)d_in[0];
    const int*   ei  = (const int*)d_in[1];
    const float* W1l = (const float*)d_in[2];
    const float* b1  = (const float*)d_in[3];
    const float* W1r = (const float*)d_in[4];
    const float* W2l = (const float*)d_in[5];
    const float* b2  = (const float*)d_in[6];
    const float* W2r = (const float*)d_in[7];

    const int N = in_sizes[0] / FEAT;      // 50000
    const int E = in_sizes[1] / 2;         // 800000
    const int ntiles = N / 16;             // 3125 (N % 16 == 0)

    float* agg = (float*)d_ws;             // N*128
    float* cnt = agg + (size_t)N * FEAT;   // N
    float* h   = cnt + N;                  // N*128

    const int* src = ei;
    const int* dst = ei + E;

    const long long nAggCnt = (long long)N * FEAT + N;
    const int scatterBlocks = (E * 32 + 255) / 256;
    const int gemmBlocks    = (ntiles + 7) / 8;

    // ---- layer 1 ----
    zero_kernel<<<2048, 256, 0, stream>>>(agg, nAggCnt);                 // agg + cnt
    scatter_kernel<<<scatterBlocks, 256, 0, stream>>>(x, src, dst, agg, cnt, E, 1);
    sage_gemm_kernel<<<gemmBlocks, 256, 0, stream>>>(agg, cnt, x, W1l, b1, W1r,
                                                     h, ntiles, 1);
    // ---- layer 2 (degrees reused) ----
    zero_kernel<<<2048, 256, 0, stream>>>(agg, (long long)N * FEAT);
    scatter_kernel<<<scatterBlocks, 256, 0, stream>>>(h, src, dst, agg, cnt, E, 0);
    sage_gemm_kernel<<<gemmBlocks, 256, 0, stream>>>(agg, cnt, h, W2l, b2, W2r,
                                                     (float*)d_out, ntiles, 0);
}